// CustomGraphConvDGL_23776938951360
// MI455X (gfx1250) — compile-verified
//
#include <hip/hip_runtime.h>
#include <hip/hip_bf16.h>
#include <stdint.h>

typedef __attribute__((ext_vector_type(2))) float v2f;
typedef __attribute__((ext_vector_type(8))) float v8f;

#define FEATS 64

// ---------------- degree pipeline ----------------

__global__ void k_zero_deg(int* __restrict__ deg, int n) {
    int i = blockIdx.x * blockDim.x + threadIdx.x;
    if (i < n) deg[i] = 0;
}

__global__ void k_count_deg(const int* __restrict__ dst, int* __restrict__ deg, int e) {
    int i = blockIdx.x * blockDim.x + threadIdx.x;
    if (i < e) atomicAdd(&deg[dst[i]], 1);
}

__global__ void k_dis(const int* __restrict__ deg, float* __restrict__ dis, int n) {
    int i = blockIdx.x * blockDim.x + threadIdx.x;
    if (i < n) dis[i] = rsqrtf((float)(deg[i] + 1));   // +1 = self loop; always > 0
}

// ---------------- h = x @ W via V_WMMA_F32_16X16X4_F32 ----------------
// Full 16-row tiles only (no per-row guards -> unconditional store clause).
// One wave owns one 16x16 output tile at a time; colTile = wave % 4 (OUT=64),
// waves grid-stride over row tiles. All 16 B-fragments (whole K for this
// column tile, 16 KB weight matrix) are preloaded into registers once per wave.

__global__ void k_gemm_wmma(const float* __restrict__ x, const float* __restrict__ w,
                            float* __restrict__ h, int nFullTiles) {
    const int lane    = threadIdx.x & 31;
    const int wave    = (blockIdx.x * blockDim.x + threadIdx.x) >> 5;
    const int nWaves  = (gridDim.x * blockDim.x) >> 5;
    const int colTile = wave & 3;
    const int group   = wave >> 2;
    const int nGroups = nWaves >> 2;
    const int laneLo  = lane & 15;
    const int laneHiK = (lane >> 4) << 1;   // lanes 0-15: K base 0; lanes 16-31: K base 2
    const int colBase = colTile << 4;

    // B fragment for k-step ks: lane holds W[k][colBase+laneLo], W[k+1][colBase+laneLo]
    // with k = ks*4 + laneHiK (mirrors the A-matrix 16x4 f32 layout).
    v2f bfrag[16];
    {
        const int col = colBase + laneLo;
        #pragma unroll
        for (int ks = 0; ks < 16; ++ks) {
            const int k = (ks << 2) + laneHiK;
            v2f b;
            b.x = w[k * FEATS + col];
            b.y = w[(k + 1) * FEATS + col];
            bfrag[ks] = b;
        }
    }

    for (int rt = group; rt < nFullTiles; rt += nGroups) {
        const int rowBase = rt << 4;
        const float* ap = x + (rowBase + laneLo) * FEATS + laneHiK;

        v8f c = {0.f, 0.f, 0.f, 0.f, 0.f, 0.f, 0.f, 0.f};
        #pragma unroll
        for (int ks = 0; ks < 16; ++ks) {
            v2f a = *(const v2f*)(ap + (ks << 2));   // global_load_b64, 8B aligned
            c = __builtin_amdgcn_wmma_f32_16x16x4_f32(
                    /*neg_a=*/false, a, /*neg_b=*/false, bfrag[ks],
                    /*c_mod=*/(short)0, c, /*reuse_a=*/false, /*reuse_b=*/false);
        }

        // C/D layout: VGPR v -> M = v + (lane>=16 ? 8 : 0), N = lane&15.
        // Unconditional stores: full tiles guaranteed.
        float* hp = h + (rowBase + ((lane >> 4) << 3)) * FEATS + colBase + laneLo;
        #pragma unroll
        for (int v = 0; v < 8; ++v) {
            hp[v * FEATS] = c[v];
        }
    }
}

// Scalar tail for rows beyond the last full 16-row tile (launched only if needed).
__global__ void k_gemm_tail(const float* __restrict__ x, const float* __restrict__ w,
                            float* __restrict__ h, int rowStart, int n) {
    const int t   = blockIdx.x * blockDim.x + threadIdx.x;
    const int row = rowStart + (t >> 6);
    if (row >= n) return;
    const int col = t & 63;
    float acc = 0.f;
    #pragma unroll 8
    for (int k = 0; k < FEATS; ++k)
        acc += x[row * FEATS + k] * w[k * FEATS + col];
    h[row * FEATS + col] = acc;
}

// ---------------- out = bias + h * dis^2  (self loop; also un-poisons d_out) -----

__global__ void k_init_out(const float* __restrict__ h, const float* __restrict__ dis,
                           const float* __restrict__ bias, float* __restrict__ out, int n) {
    const int t = blockIdx.x * blockDim.x + threadIdx.x;
    const int i = t >> 4;                  // 16 threads per node, float4 each (64 feats)
    if (i >= n) return;
    const int f = (t & 15) << 2;
    const float sc = dis[i] * dis[i];
    const float4 hv = *(const float4*)(h + i * FEATS + f);
    const float4 bv = *(const float4*)(bias + f);
    float4 o;
    o.x = bv.x + hv.x * sc;
    o.y = bv.y + hv.y * sc;
    o.z = bv.z + hv.z * sc;
    o.w = bv.w + hv.w * sc;
    *(float4*)(out + i * FEATS + f) = o;
}

// ---------------- edge scatter: out[dst] += h[src] * dis[src]*dis[dst] ----------
// 16 lanes per edge, each handling a float4 feature chunk: coalesced 256B gather
// from h[src] (L2-resident: h is 25.6MB << 192MB L2), 4 f32 atomics per lane.

__global__ void k_edge_scatter(const int* __restrict__ src, const int* __restrict__ dst,
                               const float* __restrict__ dis, const float* __restrict__ h,
                               float* __restrict__ out, int e) {
    const long long t = (long long)blockIdx.x * blockDim.x + threadIdx.x;
    const int ei = (int)(t >> 4);
    if (ei >= e) return;
    const int f = ((int)t & 15) << 2;
    const int s = src[ei];
    const int d = dst[ei];
    const float wgt = dis[s] * dis[d];
    const float4 hv = *(const float4*)(h + s * FEATS + f);
    float* o = out + d * FEATS + f;
    atomicAdd(o + 0, hv.x * wgt);
    atomicAdd(o + 1, hv.y * wgt);
    atomicAdd(o + 2, hv.z * wgt);
    atomicAdd(o + 3, hv.w * wgt);
}

// ---------------- launch ----------------

extern "C" void kernel_launch(void* const* d_in, const int* in_sizes, int n_in,
                              void* d_out, int out_size, void* d_ws, size_t ws_size,
                              hipStream_t stream) {
    const float* x    = (const float*)d_in[0];
    const int*   src  = (const int*)d_in[1];
    const int*   dst  = (const int*)d_in[2];
    const float* w    = (const float*)d_in[3];
    const float* bias = (const float*)d_in[4];
    float* out = (float*)d_out;

    const int n = in_sizes[0] / FEATS;   // 100000
    const int e = in_sizes[1];           // 1600000

    // workspace layout: dis[n] f32 | deg[n] i32 | h[n*64] f32 (16B aligned)
    float* dis = (float*)d_ws;
    int*   deg = (int*)(dis + n);
    float* h   = (float*)(((uintptr_t)(deg + n) + 15) & ~(uintptr_t)15);

    const int B = 256;
    k_zero_deg <<<(n + B - 1) / B, B, 0, stream>>>(deg, n);
    k_count_deg<<<(e + B - 1) / B, B, 0, stream>>>(dst, deg, e);
    k_dis      <<<(n + B - 1) / B, B, 0, stream>>>(deg, dis, n);

    const int nFullTiles = n / 16;       // 6250 for n=100000 (no tail)
    k_gemm_wmma<<<512, 256, 0, stream>>>(x, w, h, nFullTiles);
    const int tailRows = n - nFullTiles * 16;
    if (tailRows > 0) {
        const long long tt = (long long)tailRows * FEATS;
        k_gemm_tail<<<(int)((tt + B - 1) / B), B, 0, stream>>>(x, w, h, nFullTiles * 16, n);
    }

    const long long tInit = (long long)n * 16;
    k_init_out<<<(int)((tInit + B - 1) / B), B, 0, stream>>>(h, dis, bias, out, n);

    const long long tEdge = (long long)e * 16;
    k_edge_scatter<<<(int)((tEdge + B - 1) / B), B, 0, stream>>>(src, dst, dis, h, out, e);
}